// GroupMat_4286377361698
// MI455X (gfx1250) — compile-verified
//
#include <hip/hip_runtime.h>
#include <hip/hip_bf16.h>

// Problem constants (match reference)
#define NN 8192
#define EE 262144          // = 2^18
#define RR 2
#define FF 111
#define DD 32
#define HH 256
#define CC 15

typedef __attribute__((ext_vector_type(2))) float v2f;
typedef __attribute__((ext_vector_type(8))) float v8f;

// f32 WMMA: D(16x16 f32) = A(16x4 f32) * B(4x16 f32) + C
__device__ __forceinline__ v8f wmma4(v2f a, v2f b, v8f c) {
    return __builtin_amdgcn_wmma_f32_16x16x4_f32(
        /*neg_a=*/false, a, /*neg_b=*/false, b,
        /*c_mod=*/(short)0, c, /*reuse_a=*/false, /*reuse_b=*/false);
}

// ---------------------------------------------------------------------------
// Wcat[96,256] = [Wself0+Wself1 ; Wneigh0 ; Wneigh1], bsum[256] = b0+b1
__global__ void prep_wcat(const float* __restrict__ Wself,
                          const float* __restrict__ Wneigh,
                          const float* __restrict__ bgnn,
                          float* __restrict__ Wcat, float* __restrict__ bsum) {
    int idx = blockIdx.x * blockDim.x + threadIdx.x;
    if (idx < 96 * 256) {
        int k = idx >> 8, n = idx & 255;
        float v;
        if (k < 32)       v = Wself[k * 256 + n] + Wself[32 * 256 + k * 256 + n];
        else if (k < 64)  v = Wneigh[(k - 32) * 256 + n];
        else              v = Wneigh[32 * 256 + (k - 64) * 256 + n];
        Wcat[idx] = v;
    }
    if (idx < 256) bsum[idx] = bgnn[idx] + bgnn[256 + idx];
}

// ---------------------------------------------------------------------------
// X96[:,0:32] = x_note[8192,111] @ W_embed[111,32] + b_embed  (row stride 96)
__global__ void gemm_embed(const float* __restrict__ A, const float* __restrict__ B,
                           const float* __restrict__ bias, float* __restrict__ X96) {
    int gid  = blockIdx.x * blockDim.x + threadIdx.x;
    int wave = gid >> 5, lane = threadIdx.x & 31;
    int tileN = wave & 1, tileM = wave >> 1;            // 512 x 2 tiles
    int row  = tileM * 16 + (lane & 15);
    int col  = tileN * 16 + (lane & 15);
    int kofs = (lane >> 4) * 2;
    v8f acc = {0.f,0.f,0.f,0.f,0.f,0.f,0.f,0.f};
    for (int k = 0; k < 112; k += 4) {
        int kk = k + kofs;
        v2f a, b;
        a.x = (kk     < FF) ? A[row * FF + kk]     : 0.f;
        a.y = (kk + 1 < FF) ? A[row * FF + kk + 1] : 0.f;
        b.x = (kk     < FF) ? B[kk * DD + col]       : 0.f;
        b.y = (kk + 1 < FF) ? B[(kk + 1) * DD + col] : 0.f;
        acc = wmma4(a, b, acc);
    }
    int mbase = tileM * 16 + ((lane >> 4) ? 8 : 0);
    float bv = bias[col];
#pragma unroll
    for (int v = 0; v < 8; ++v)
        X96[(mbase + v) * 96 + col] = acc[v] + bv;
}

// ---------------------------------------------------------------------------
// Per-edge mean-agg numerators + counts. Reads X96 cols 0..31, writes 32..95.
__global__ void aggregate(const int* __restrict__ edges, float* __restrict__ X96,
                          float* __restrict__ counts) {
    int tid  = blockIdx.x * blockDim.x + threadIdx.x;   // 2*E*32 threads
    int lane = tid & 31;
    int ei   = tid >> 5;                                 // edge index over 2E
    int r    = ei >> 18;                                 // E = 2^18
    int e    = ei & (EE - 1);
    int src  = edges[r * 2 * EE + e];
    int dst  = edges[r * 2 * EE + EE + e];
    atomicAdd(&X96[dst * 96 + 32 + r * 32 + lane], X96[src * 96 + lane]);
    if (lane == 0) atomicAdd(&counts[r * NN + dst], 1.0f);
}

// mean = sum / max(cnt,1)
__global__ void mean_div(float* __restrict__ X96, const float* __restrict__ counts) {
    int tid = blockIdx.x * blockDim.x + threadIdx.x;     // 8192*64
    int n = tid >> 6, q = tid & 63, r = q >> 5, f = q & 31;
    float c = counts[r * NN + n];
    if (c < 1.f) c = 1.f;
    X96[n * 96 + 32 + r * 32 + f] /= c;
}

// ---------------------------------------------------------------------------
// H[8192,256] = relu(X96[8192,96] @ Wcat[96,256] + bsum)
__global__ void gemm_main(const float* __restrict__ X96, const float* __restrict__ Wcat,
                          const float* __restrict__ bsum, float* __restrict__ H) {
    int gid  = blockIdx.x * blockDim.x + threadIdx.x;
    int wave = gid >> 5, lane = threadIdx.x & 31;
    int tileN = wave & 15, tileM = wave >> 4;            // 512 x 16 tiles
    int row  = tileM * 16 + (lane & 15);
    int col  = tileN * 16 + (lane & 15);
    int kofs = (lane >> 4) * 2;
    const float* arow = X96 + row * 96 + kofs;
    v8f acc = {0.f,0.f,0.f,0.f,0.f,0.f,0.f,0.f};
#pragma unroll
    for (int k = 0; k < 96; k += 4) {
        v2f a = *(const v2f*)(arow + k);                 // 8B aligned: even index
        v2f b;
        b.x = Wcat[(k + kofs) * 256 + col];
        b.y = Wcat[(k + kofs + 1) * 256 + col];
        acc = wmma4(a, b, acc);
    }
    int mbase = tileM * 16 + ((lane >> 4) ? 8 : 0);
    float bv = bsum[col];
#pragma unroll
    for (int v = 0; v < 8; ++v) {
        float d = acc[v] + bv;
        H[(mbase + v) * 256 + col] = d > 0.f ? d : 0.f;
    }
}

// ---------------------------------------------------------------------------
// y[8192,16] = H[8192,256] @ W_s1[256,15]  (col 15 zero-padded)
__global__ void gemm_ws1(const float* __restrict__ H, const float* __restrict__ Ws,
                         float* __restrict__ y) {
    int gid  = blockIdx.x * blockDim.x + threadIdx.x;
    int wave = gid >> 5, lane = threadIdx.x & 31;
    int tileM = wave;                                    // 512 tiles, single N tile
    int row  = tileM * 16 + (lane & 15);
    int col  = lane & 15;
    int kofs = (lane >> 4) * 2;
    bool ok  = col < CC;
    const float* arow = H + row * 256 + kofs;
    v8f acc = {0.f,0.f,0.f,0.f,0.f,0.f,0.f,0.f};
    for (int k = 0; k < 256; k += 4) {
        v2f a = *(const v2f*)(arow + k);
        v2f b;
        b.x = ok ? Ws[(k + kofs) * CC + col]     : 0.f;
        b.y = ok ? Ws[(k + kofs + 1) * CC + col] : 0.f;
        acc = wmma4(a, b, acc);
    }
    int mbase = tileM * 16 + ((lane >> 4) ? 8 : 0);
#pragma unroll
    for (int v = 0; v < 8; ++v)
        y[(mbase + v) * 16 + col] = acc[v];              // col15 accumulates zeros
}

// ---------------------------------------------------------------------------
// z[src,:] += y[dst,:]   (a_sum @ y via edge list; both relations)
__global__ void scatter_z(const int* __restrict__ edges, const float* __restrict__ y,
                          float* __restrict__ z) {
    int tid = blockIdx.x * blockDim.x + threadIdx.x;     // 2*E*16
    int c   = tid & 15;
    int ei  = tid >> 4;
    int r   = ei >> 18;
    int e   = ei & (EE - 1);
    if (c < CC) {
        int src = edges[r * 2 * EE + e];
        int dst = edges[r * 2 * EE + EE + e];
        atomicAdd(&z[src * 16 + c], y[dst * 16 + c]);
    }
}

// row softmax over 15 cols -> S1p (padded) and d_out S1 region
__global__ void softmax1(const float* __restrict__ z, float* __restrict__ S1p,
                         float* __restrict__ outS1) {
    int n = blockIdx.x * blockDim.x + threadIdx.x;
    if (n >= NN) return;
    float v[CC], m = -1e30f;
#pragma unroll
    for (int c = 0; c < CC; ++c) { v[c] = z[n * 16 + c]; m = fmaxf(m, v[c]); }
    float s = 0.f;
#pragma unroll
    for (int c = 0; c < CC; ++c) { v[c] = __expf(v[c] - m); s += v[c]; }
    float inv = 1.f / s;
#pragma unroll
    for (int c = 0; c < CC; ++c) {
        float p = v[c] * inv;
        S1p[n * 16 + c] = p;
        outS1[n * CC + c] = p;
    }
    S1p[n * 16 + 15] = 0.f;
}

// ---------------------------------------------------------------------------
// x_p[15,256] = S1^T @ H  (M=16 padded, K=8192 split 16 ways, atomic reduce)
__global__ void gemm_xp(const float* __restrict__ S1p, const float* __restrict__ H,
                        float* __restrict__ x_p) {
    int gid  = blockIdx.x * blockDim.x + threadIdx.x;
    int wave = gid >> 5, lane = threadIdx.x & 31;
    int tileN  = wave & 15;
    int ksplit = wave >> 4;                               // 16 K-slices of 512
    int colb = tileN * 16;
    int col  = colb + (lane & 15);
    int kofs = (lane >> 4) * 2;
    int k0 = ksplit * 512, k1 = k0 + 512;
    v8f acc = {0.f,0.f,0.f,0.f,0.f,0.f,0.f,0.f};
    for (int k = k0; k < k1; k += 4) {
        int kk = k + kofs;
        v2f a, b;
        a.x = S1p[kk * 16 + (lane & 15)];                 // A[m=c, k=n] = S1[n,c]
        a.y = S1p[(kk + 1) * 16 + (lane & 15)];
        b.x = H[kk * 256 + col];
        b.y = H[(kk + 1) * 256 + col];
        acc = wmma4(a, b, acc);
    }
    int mbase = (lane >> 4) ? 8 : 0;
#pragma unroll
    for (int v = 0; v < 8; ++v) {
        int row = mbase + v;
        if (row < CC) atomicAdd(&x_p[row * 256 + col], acc[v]);
    }
}

// ---------------------------------------------------------------------------
// adj_p[r] = S1^T A_r S1 = sum over edges of outer(S1[src], S1[dst])
__global__ void adjp_kernel(const int* __restrict__ edges, const float* __restrict__ S1p,
                            float* __restrict__ adj_p) {
    __shared__ float acc[CC * 16];
    int blocksPerRel = gridDim.x >> 1;
    int r = blockIdx.x / blocksPerRel;
    int b = blockIdx.x % blocksPerRel;
    for (int i = threadIdx.x; i < CC * 16; i += blockDim.x) acc[i] = 0.f;
    __syncthreads();
    const int* src = edges + r * 2 * EE;
    const int* dst = src + EE;
    for (int e = b * blockDim.x + threadIdx.x; e < EE; e += blocksPerRel * blockDim.x) {
        int s = src[e], d = dst[e];
        float sv[CC], dv[CC];
#pragma unroll
        for (int i = 0; i < CC; ++i) { sv[i] = S1p[s * 16 + i]; dv[i] = S1p[d * 16 + i]; }
#pragma unroll
        for (int i = 0; i < CC; ++i)
#pragma unroll
            for (int j = 0; j < CC; ++j)
                atomicAdd(&acc[i * 16 + j], sv[i] * dv[j]);
    }
    __syncthreads();
    for (int i = threadIdx.x; i < CC * 16; i += blockDim.x) {
        int ii = i >> 4, jj = i & 15;
        if (jj < CC) atomicAdd(&adj_p[r * CC * CC + ii * CC + jj], acc[i]);
    }
}

// ---------------------------------------------------------------------------
// cluster2 tail: a2=sum_r adj_p; t=a2@x_p; z2=t@W_s2; S2=softmax; x2=S2^T@x_p
__global__ void cluster2_kernel(const float* __restrict__ x_p, const float* __restrict__ adj_p,
                                const float* __restrict__ Ws2,
                                float* __restrict__ out_x, float* __restrict__ out_S2) {
    __shared__ float a2[CC * CC];
    __shared__ float t[CC * 256];
    __shared__ float z2[CC * CC];
    __shared__ float S2[CC * CC];
    int tid = threadIdx.x;
    if (tid < CC * CC) a2[tid] = adj_p[tid] + adj_p[CC * CC + tid];
    __syncthreads();
    for (int idx = tid; idx < CC * 256; idx += blockDim.x) {
        int i = idx >> 8, h = idx & 255;
        float s = 0.f;
#pragma unroll
        for (int j = 0; j < CC; ++j) s += a2[i * CC + j] * x_p[j * 256 + h];
        t[idx] = s;
    }
    __syncthreads();
    for (int idx = tid; idx < CC * CC; idx += blockDim.x) {
        int i = idx / CC, c = idx % CC;
        float s = 0.f;
        for (int h = 0; h < 256; ++h) s += t[i * 256 + h] * Ws2[h * CC + c];
        z2[idx] = s;
    }
    __syncthreads();
    if (tid < CC) {
        float m = -1e30f;
#pragma unroll
        for (int c = 0; c < CC; ++c) m = fmaxf(m, z2[tid * CC + c]);
        float ex[CC], s = 0.f;
#pragma unroll
        for (int c = 0; c < CC; ++c) { ex[c] = __expf(z2[tid * CC + c] - m); s += ex[c]; }
        float inv = 1.f / s;
#pragma unroll
        for (int c = 0; c < CC; ++c) {
            float p = ex[c] * inv;
            S2[tid * CC + c] = p;
            out_S2[tid * CC + c] = p;
        }
    }
    __syncthreads();
    for (int idx = tid; idx < CC * 256; idx += blockDim.x) {
        int c = idx >> 8, h = idx & 255;
        float s = 0.f;
#pragma unroll
        for (int i = 0; i < CC; ++i) s += S2[i * CC + c] * x_p[i * 256 + h];
        out_x[c * 256 + h] = s;
    }
}

// ---------------------------------------------------------------------------
extern "C" void kernel_launch(void* const* d_in, const int* in_sizes, int n_in,
                              void* d_out, int out_size, void* d_ws, size_t ws_size,
                              hipStream_t stream) {
    const float* x_note  = (const float*)d_in[0];
    const int*   edges   = (const int*)  d_in[1];
    const float* W_embed = (const float*)d_in[2];
    const float* b_embed = (const float*)d_in[3];
    const float* W_self  = (const float*)d_in[4];
    const float* W_neigh = (const float*)d_in[5];
    const float* b_gnn   = (const float*)d_in[6];
    const float* W_s1    = (const float*)d_in[7];
    const float* W_s2    = (const float*)d_in[8];
    float* out = (float*)d_out;          // [x(15*256) | S1(8192*15) | S2(15*15)]

    float* ws     = (float*)d_ws;
    float* X96    = ws;                      // 8192*96
    float* counts = X96    + NN * 96;        // 2*8192
    float* Wcat   = counts + 2 * NN;         // 96*256
    float* bsum   = Wcat   + 96 * 256;       // 256
    float* H      = bsum   + 256;            // 8192*256
    float* y      = H      + NN * 256;       // 8192*16
    float* z      = y      + NN * 16;        // 8192*16
    float* S1p    = z      + NN * 16;        // 8192*16
    float* x_p    = S1p    + NN * 16;        // 16*256
    float* adj_p  = x_p    + 16 * 256;       // 2*15*15 (+pad)

    // zero accumulators (capturable memset nodes)
    hipMemsetAsync(X96,    0, (size_t)NN * 96 * 4, stream);
    hipMemsetAsync(counts, 0, (size_t)2 * NN * 4, stream);
    hipMemsetAsync(z,      0, (size_t)NN * 16 * 4, stream);
    hipMemsetAsync(x_p,    0, (size_t)(16 * 256 + 512) * 4, stream);

    prep_wcat <<<(96 * 256 + 255) / 256, 256, 0, stream>>>(W_self, W_neigh, b_gnn, Wcat, bsum);
    gemm_embed<<<128,   256, 0, stream>>>(x_note, W_embed, b_embed, X96);     // 1024 waves
    aggregate <<<65536, 256, 0, stream>>>(edges, X96, counts);                // 2E*32 threads
    mean_div  <<<2048,  256, 0, stream>>>(X96, counts);
    gemm_main <<<1024,  256, 0, stream>>>(X96, Wcat, bsum, H);                // 8192 waves
    gemm_ws1  <<<64,    256, 0, stream>>>(H, W_s1, y);                        // 512 waves
    scatter_z <<<32768, 256, 0, stream>>>(edges, y, z);                       // 2E*16 threads
    softmax1  <<<32,    256, 0, stream>>>(z, S1p, out + CC * 256);
    gemm_xp   <<<32,    256, 0, stream>>>(S1p, H, x_p);                       // 256 waves
    adjp_kernel<<<128,  256, 0, stream>>>(edges, S1p, adj_p);
    cluster2_kernel<<<1, 256, 0, stream>>>(x_p, adj_p, W_s2,
                                           out, out + CC * 256 + NN * CC);
}